// GNNSage_43937515438448
// MI455X (gfx1250) — compile-verified
//
#include <hip/hip_runtime.h>

typedef __attribute__((ext_vector_type(2))) float v2f;
typedef __attribute__((ext_vector_type(8))) float v8f;

#define FEAT 128
#define CLS  2
#define LDSP 132    // padded LDS row (floats) -> conflict-free column reads
#define MT   4      // M-tiles (of 16 rows) per block
#define MROWS (MT * 16)

// ---------------- K1: degree (float, matches jnp ones segment_sum) ----------
__global__ void deg_kernel(const int* __restrict__ dst, float* __restrict__ deg, int E) {
  int e = blockIdx.x * blockDim.x + threadIdx.x;
  if (e < E) atomicAdd(&deg[dst[e]], 1.0f);
}

// ---------------- K2: layer-1 scatter: agg1[dst] += features[src] -----------
// one wave per edge: 32 lanes x float4 = 128 floats; atomics resolve in L2
__global__ void scatter1_kernel(const float* __restrict__ feat,
                                const int* __restrict__ src,
                                const int* __restrict__ dst,
                                float* agg, int E) {
  long g = (long)blockIdx.x * blockDim.x + threadIdx.x;
  int e = (int)(g >> 5);
  int lane = (int)(g & 31);
  if (e >= E) return;
  int s = src[e], d = dst[e];
  float4 v = *(const float4*)(feat + (long)s * FEAT + lane * 4);
  float* ar = agg + (long)d * FEAT + lane * 4;
  atomicAdd(ar + 0, v.x);
  atomicAdd(ar + 1, v.y);
  atomicAdd(ar + 2, v.z);
  atomicAdd(ar + 3, v.w);
}

// ---------------- K3: layer-1 fused GEMM + ReLU via V_WMMA_F32_16X16X4_F32 --
// h1 = relu(X @ Wself + (agg/deg) @ Wneigh + b)
// Block = 64 nodes x 128 cols; 8 waves each own a 16-col slice; each B
// fragment is reused across 4 M-tiles (4x less L2 weight traffic).
// h1 may alias agg (rows staged to LDS before being overwritten).
__global__ __launch_bounds__(256) void layer1_kernel(
    const float* __restrict__ feat, const float* agg,
    const float* __restrict__ deg,
    const float* __restrict__ Wself, const float* __restrict__ Wneigh,
    const float* __restrict__ bias, float* h1, int N) {
  __shared__ float As[MROWS * LDSP];  // features tile   [64][128]
  __shared__ float Am[MROWS * LDSP];  // mean-neigh tile [64][128]

  const int tid  = threadIdx.x;
  const int base = blockIdx.x * MROWS;

  // stage A tiles: MROWS*32 float4 chunks; zero-fill rows past N
  for (int q = tid; q < MROWS * 32; q += 256) {
    int row = q >> 5;
    int c4  = (q & 31) * 4;
    int gr  = base + row;
    float4 f = make_float4(0.f, 0.f, 0.f, 0.f);
    float4 a = make_float4(0.f, 0.f, 0.f, 0.f);
    float inv = 0.f;
    if (gr < N) {
      long goff = (long)gr * FEAT + c4;
      f = *(const float4*)(feat + goff);
      a = *(const float4*)(agg + goff);
      inv = 1.0f / fmaxf(deg[gr], 1.0f);
    }
    float* ps = &As[row * LDSP + c4];
    float* pm = &Am[row * LDSP + c4];
    ps[0] = f.x; ps[1] = f.y; ps[2] = f.z; ps[3] = f.w;
    pm[0] = a.x * inv; pm[1] = a.y * inv; pm[2] = a.z * inv; pm[3] = a.w * inv;
  }
  __syncthreads();

  const int wid  = tid >> 5;        // N-tile select: cols [wid*16, wid*16+16)
  const int lane = tid & 31;
  const int r    = lane & 15;       // A row-in-tile / B,C column-in-tile
  const int hi   = lane >> 4;       // K phase within 16x16x4 op
  const int col  = wid * 16 + r;    // global output feature

  // C init: bias depends only on N -> broadcast into all acc VGPRs
  v8f c[MT];
  const float bv = bias[col];
#pragma unroll
  for (int t = 0; t < MT; ++t)
#pragma unroll
    for (int i = 0; i < 8; ++i) c[t][i] = bv;

#pragma unroll
  for (int k0 = 0; k0 < FEAT; k0 += 4) {
    const int ka = k0 + 2 * hi;
    v2f bs, bn;
    bs.x = Wself[ka * FEAT + col];   bs.y = Wself[(ka + 1) * FEAT + col];
    bn.x = Wneigh[ka * FEAT + col];  bn.y = Wneigh[(ka + 1) * FEAT + col];
#pragma unroll
    for (int t = 0; t < MT; ++t) {
      const int rr = t * 16 + r;
      v2f a;
      a.x = As[rr * LDSP + ka];  a.y = As[rr * LDSP + ka + 1];
      c[t] = __builtin_amdgcn_wmma_f32_16x16x4_f32(false, a, false, bs,
                                                   (short)0, c[t], false, false);
      a.x = Am[rr * LDSP + ka];  a.y = Am[rr * LDSP + ka + 1];
      c[t] = __builtin_amdgcn_wmma_f32_16x16x4_f32(false, a, false, bn,
                                                   (short)0, c[t], false, false);
    }
  }

  // C layout: VGPR i -> M = t*16 + i + 8*hi, N = col ; fuse ReLU
#pragma unroll
  for (int t = 0; t < MT; ++t) {
#pragma unroll
    for (int i = 0; i < 8; ++i) {
      int m = base + t * 16 + i + 8 * hi;
      if (m < N) h1[(long)m * FEAT + col] = fmaxf(c[t][i], 0.0f);
    }
  }
}

// ---------------- K4: project-before-aggregate for layer 2 ------------------
// one wave per node: out_self = h1@Wself2 + b2 (to d_out), t2 = h1@Wneigh2
__global__ __launch_bounds__(256) void layer2_gemm_kernel(
    const float* __restrict__ h1, const float* __restrict__ Wself,
    const float* __restrict__ Wneigh, const float* __restrict__ bias,
    float* __restrict__ out, float* __restrict__ t2, int N) {
  const int wid  = threadIdx.x >> 5;
  const int lane = threadIdx.x & 31;
  const int node = blockIdx.x * 8 + wid;
  if (node >= N) return;

  float4 h = *(const float4*)(h1 + (long)node * FEAT + lane * 4);
  // W rows lane*4..lane*4+3, 2 cols each -> 8 contiguous floats
  float4 w0 = *(const float4*)(Wself + lane * 8);
  float4 w1 = *(const float4*)(Wself + lane * 8 + 4);
  float s0 = h.x * w0.x + h.y * w0.z + h.z * w1.x + h.w * w1.z;
  float s1 = h.x * w0.y + h.y * w0.w + h.z * w1.y + h.w * w1.w;
  float4 n0 = *(const float4*)(Wneigh + lane * 8);
  float4 n1 = *(const float4*)(Wneigh + lane * 8 + 4);
  float t0 = h.x * n0.x + h.y * n0.z + h.z * n1.x + h.w * n1.z;
  float t1 = h.x * n0.y + h.y * n0.w + h.z * n1.y + h.w * n1.w;

#pragma unroll
  for (int m = 16; m >= 1; m >>= 1) {
    s0 += __shfl_xor(s0, m, 32);
    s1 += __shfl_xor(s1, m, 32);
    t0 += __shfl_xor(t0, m, 32);
    t1 += __shfl_xor(t1, m, 32);
  }
  if (lane == 0) {
    out[node * CLS + 0] = s0 + bias[0];
    out[node * CLS + 1] = s1 + bias[1];
    t2[node * CLS + 0]  = t0;
    t2[node * CLS + 1]  = t1;
  }
}

// ---------------- K5: layer-2 scatter (2 floats/edge, 64x less traffic) -----
__global__ void scatter2_kernel(const float* __restrict__ t2,
                                const int* __restrict__ src,
                                const int* __restrict__ dst,
                                float* __restrict__ agg2, int E) {
  int e = blockIdx.x * blockDim.x + threadIdx.x;
  if (e >= E) return;
  int s = src[e], d = dst[e];
  float2 t = *(const float2*)(t2 + (long)s * CLS);
  atomicAdd(&agg2[d * CLS + 0], t.x);
  atomicAdd(&agg2[d * CLS + 1], t.y);
}

// ---------------- K6: out += agg2 / max(deg,1) ------------------------------
__global__ void finalize_kernel(const float* __restrict__ agg2,
                                const float* __restrict__ deg,
                                float* __restrict__ out, int N) {
  int i = blockIdx.x * blockDim.x + threadIdx.x;
  if (i >= N) return;
  float inv = 1.0f / fmaxf(deg[i], 1.0f);
  out[i * CLS + 0] += agg2[i * CLS + 0] * inv;
  out[i * CLS + 1] += agg2[i * CLS + 1] * inv;
}

extern "C" void kernel_launch(void* const* d_in, const int* in_sizes, int n_in,
                              void* d_out, int out_size, void* d_ws, size_t ws_size,
                              hipStream_t stream) {
  const float* feat = (const float*)d_in[0];
  const int*   src  = (const int*)d_in[1];
  const int*   dst  = (const int*)d_in[2];
  const float* Ws1  = (const float*)d_in[3];
  const float* Wn1  = (const float*)d_in[4];
  const float* b1   = (const float*)d_in[5];
  const float* Ws2  = (const float*)d_in[6];
  const float* Wn2  = (const float*)d_in[7];
  const float* b2   = (const float*)d_in[8];
  float* out = (float*)d_out;

  const int E = in_sizes[1];
  const int N = in_sizes[0] / FEAT;  // 100000

  // workspace (floats): deg[N] | buf1[N*128] (agg1 then h1) | agg2[N*2] | t2[N*2]
  float* ws   = (float*)d_ws;
  float* deg  = ws;
  float* buf1 = deg + N;
  float* agg2 = buf1 + (size_t)N * FEAT;
  float* t2   = agg2 + (size_t)N * CLS;

  // zero deg + agg1 + agg2 in one contiguous memset (graph-capturable)
  hipMemsetAsync(deg, 0, (size_t)N * (1 + FEAT + CLS) * sizeof(float), stream);

  deg_kernel<<<(E + 255) / 256, 256, 0, stream>>>(dst, deg, E);

  long tthr = (long)E * 32;
  scatter1_kernel<<<(int)((tthr + 255) / 256), 256, 0, stream>>>(feat, src, dst, buf1, E);

  layer1_kernel<<<(N + MROWS - 1) / MROWS, 256, 0, stream>>>(feat, buf1, deg,
                                                             Ws1, Wn1, b1, buf1, N);

  layer2_gemm_kernel<<<(N + 7) / 8, 256, 0, stream>>>(buf1, Ws2, Wn2, b2, out, t2, N);

  scatter2_kernel<<<(E + 255) / 256, 256, 0, stream>>>(t2, src, dst, agg2, E);

  finalize_kernel<<<(N + 255) / 256, 256, 0, stream>>>(agg2, deg, out, N);
}